// RoIHeads_8830452760632
// MI455X (gfx1250) — compile-verified
//
#include <hip/hip_runtime.h>
#include <hip/hip_bf16.h>
#include <math.h>

// ---------------------------------------------------------------------------
// RoIHeads postprocess (Faster R-CNN box head) for MI455X / gfx1250.
//
// ~10k candidates, ~400KB state: latency bound, single-WGP problem. No matrix
// structure -> WMMA inapplicable. CDNA5-specific paths used:
//   * global_load_async_to_lds_b128 (+ s_wait_asynccnt) to stage the masked
//     score array into LDS, where it stays resident across the 100-iteration
//     greedy argmax+suppress NMS loop (equivalent to sorted greedy NMS,
//     avoiding the reference's full sort and 10^8-entry IoU matrix).
//   * wave32 shuffle argmax: 2 barriers/iteration instead of a 10-level LDS
//     tree, minimizing the serial-latency term that dominates this kernel.
// ---------------------------------------------------------------------------

#define N_PROP      1000
#define NUM_CLASSES 11
#define M_CAND      (N_PROP * (NUM_CLASSES - 1))   // 10000 candidates
#define DET_PER_IMG 100
#define IMG_W       800.0f
#define IMG_H       800.0f
#define SCORE_TH    0.05f
#define NMS_TH      0.5f
#define MIN_SIZE    0.01f
#define BBOX_CLIP   4.135166556742356f              // log(1000/16)
#define NMS_THREADS 1024                            // 32 wave32 on one WGP
#define NUM_WAVES   (NMS_THREADS / 32)

// ---------------------------------------------------------------------------
// Kernel 1: decode boxes + softmax scores + validity mask.
// One thread per (proposal, class>=1) candidate. m = p*10 + (c-1), c = label.
// ---------------------------------------------------------------------------
__global__ void roih_decode_kernel(const float* __restrict__ logits,   // [1000,11]
                                   const float* __restrict__ reg,      // [1000,44]
                                   const float* __restrict__ props,    // [1000,4]
                                   float* __restrict__ ms,             // [10000] masked score
                                   float* __restrict__ nb,             // [10000,4] offset boxes
                                   float* __restrict__ raw)            // [10000,4] raw boxes
{
    int m = blockIdx.x * blockDim.x + threadIdx.x;
    if (m >= M_CAND) return;
    int p = m / (NUM_CLASSES - 1);
    int c = m % (NUM_CLASSES - 1) + 1;     // class label 1..10

    // softmax over this proposal's 11 logits (max-subtracted)
    const float* L = logits + p * NUM_CLASSES;
    float mx = L[0];
#pragma unroll
    for (int i = 1; i < NUM_CLASSES; ++i) mx = fmaxf(mx, L[i]);
    float sum = 0.0f;
#pragma unroll
    for (int i = 0; i < NUM_CLASSES; ++i) sum += expf(L[i] - mx);
    float score = expf(L[c] - mx) / sum;

    // proposal geometry
    float px1 = props[p * 4 + 0], py1 = props[p * 4 + 1];
    float px2 = props[p * 4 + 2], py2 = props[p * 4 + 3];
    float pw = px2 - px1, ph = py2 - py1;
    float pcx = px1 + 0.5f * pw, pcy = py1 + 0.5f * ph;

    // box regression decode (weights 10,10,5,5; dw/dh clipped)
    const float* r = reg + p * (NUM_CLASSES * 4) + c * 4;
    float dx = r[0] * 0.1f;
    float dy = r[1] * 0.1f;
    float dw = fminf(r[2] * 0.2f, BBOX_CLIP);
    float dh = fminf(r[3] * 0.2f, BBOX_CLIP);
    float cx = dx * pw + pcx;
    float cy = dy * ph + pcy;
    float w  = expf(dw) * pw;
    float h  = expf(dh) * ph;

    float bx1 = fminf(fmaxf(cx - 0.5f * w, 0.0f), IMG_W);
    float by1 = fminf(fmaxf(cy - 0.5f * h, 0.0f), IMG_H);
    float bx2 = fminf(fmaxf(cx + 0.5f * w, 0.0f), IMG_W);
    float by2 = fminf(fmaxf(cy + 0.5f * h, 0.0f), IMG_H);

    bool valid = (score > SCORE_TH) &&
                 (bx2 - bx1 >= MIN_SIZE) && (by2 - by1 >= MIN_SIZE);

    float off = (float)c * (IMG_W + 1.0f);   // per-class NMS offset (801)

    raw[m * 4 + 0] = bx1;  raw[m * 4 + 1] = by1;
    raw[m * 4 + 2] = bx2;  raw[m * 4 + 3] = by2;
    nb[m * 4 + 0]  = bx1 + off;  nb[m * 4 + 1] = by1 + off;
    nb[m * 4 + 2]  = bx2 + off;  nb[m * 4 + 3] = by2 + off;
    ms[m] = valid ? score : -INFINITY;
}

// argmax-combine with stable-sort tie-break (equal score -> smaller index)
__device__ __forceinline__ void amax_combine(float& bv, int& bi, float ov, int oi) {
    if (oi >= 0 && (ov > bv || (ov == bv && (bi < 0 || oi < bi)))) {
        bv = ov; bi = oi;
    }
}

// ---------------------------------------------------------------------------
// Kernel 2: greedy NMS + top-100 selection. Single workgroup (32 wave32).
// Equivalent to sorted greedy NMS: repeatedly argmax over live masked scores
// (tie -> smallest flat index, matching stable argsort), emit, suppress.
// ---------------------------------------------------------------------------
__global__ void __launch_bounds__(NMS_THREADS)
roih_nms_kernel(const float* __restrict__ ms,
                const float* __restrict__ nb,
                const float* __restrict__ raw,
                float* __restrict__ out)     // [100*4 boxes | 100 scores | 100 labels]
{
    __shared__ alignas(16) float s_score[M_CAND];   // 40 KB live scores
    __shared__ float s_rv[NUM_WAVES];
    __shared__ int   s_ri[NUM_WAVES];
    __shared__ float s_box[5];                      // chosen offset box + area
    __shared__ int   s_done;

    const int tid = threadIdx.x;

    // zero outputs (reference pads with zeros past the kept count)
    for (int i = tid; i < DET_PER_IMG * 6; i += NMS_THREADS) out[i] = 0.0f;

    // ---- CDNA5 async stage: global -> LDS, 128b granules, ASYNCcnt ----
    for (int j = tid * 4; j < M_CAND; j += NMS_THREADS * 4) {
        unsigned lds_off = (unsigned)(uintptr_t)(&s_score[j]);   // low 32b = LDS byte addr
        unsigned long long gaddr = (unsigned long long)(uintptr_t)(ms + j);
        asm volatile("global_load_async_to_lds_b128 %0, %1, off"
                     :: "v"(lds_off), "v"(gaddr)
                     : "memory");
    }
    asm volatile("s_wait_asynccnt 0" ::: "memory");
    __syncthreads();

    const float4* nb4  = (const float4*)nb;
    const float4* raw4 = (const float4*)raw;

    const int wave = tid >> 5;
    const int lane = tid & 31;

    for (int k = 0; k < DET_PER_IMG; ++k) {
        // ---- parallel argmax over live scores ----
        float bv = -INFINITY;
        int   bi = -1;
        for (int j = tid; j < M_CAND; j += NMS_THREADS) {
            float v = s_score[j];
            if (v > bv) { bv = v; bi = j; }
        }
        // stage 1: reduce within each wave32
#pragma unroll
        for (int d = 16; d > 0; d >>= 1) {
            float ov = __shfl_down(bv, d, 32);
            int   oi = __shfl_down(bi, d, 32);
            amax_combine(bv, bi, ov, oi);
        }
        if (lane == 0) { s_rv[wave] = bv; s_ri[wave] = bi; }
        __syncthreads();
        // stage 2: first wave reduces the 32 per-wave partials
        if (tid < 32) {
            bv = s_rv[tid]; bi = s_ri[tid];
#pragma unroll
            for (int d = 16; d > 0; d >>= 1) {
                float ov = __shfl_down(bv, d, 32);
                int   oi = __shfl_down(bi, d, 32);
                amax_combine(bv, bi, ov, oi);
            }
            if (tid == 0) {
                if (bi < 0 || !(bv > -INFINITY)) {
                    s_done = 1;
                } else {
                    s_done = 0;
                    // emit: raw clipped box, raw score (== masked for kept), label
                    float4 rb = raw4[bi];
                    out[k * 4 + 0] = rb.x;  out[k * 4 + 1] = rb.y;
                    out[k * 4 + 2] = rb.z;  out[k * 4 + 3] = rb.w;
                    out[DET_PER_IMG * 4 + k] = bv;
                    out[DET_PER_IMG * 5 + k] = (float)(bi % (NUM_CLASSES - 1) + 1);
                    // broadcast chosen offset box + area for suppression sweep
                    float4 ob = nb4[bi];
                    s_box[0] = ob.x; s_box[1] = ob.y;
                    s_box[2] = ob.z; s_box[3] = ob.w;
                    s_box[4] = (ob.z - ob.x) * (ob.w - ob.y);
                    s_score[bi] = -INFINITY;   // picked: remove from live set
                }
            }
        }
        __syncthreads();
        if (s_done) break;

        // ---- parallel suppression sweep (picked box removed above) ----
        float x1 = s_box[0], y1 = s_box[1], x2 = s_box[2], y2 = s_box[3];
        float ai = s_box[4];
        for (int j = tid; j < M_CAND; j += NMS_THREADS) {
            float v = s_score[j];
            if (v > -INFINITY) {
                float4 jb = nb4[j];
                float iw = fmaxf(0.0f, fminf(x2, jb.z) - fmaxf(x1, jb.x));
                float ih = fmaxf(0.0f, fminf(y2, jb.w) - fmaxf(y1, jb.y));
                float inter = iw * ih;
                float aj = (jb.z - jb.x) * (jb.w - jb.y);
                float iou = inter / (ai + aj - inter + 1e-9f);
                if (iou > NMS_TH) s_score[j] = -INFINITY;
            }
        }
        __syncthreads();
    }
}

// ---------------------------------------------------------------------------
extern "C" void kernel_launch(void* const* d_in, const int* in_sizes, int n_in,
                              void* d_out, int out_size, void* d_ws, size_t ws_size,
                              hipStream_t stream) {
    (void)in_sizes; (void)n_in; (void)out_size; (void)ws_size;
    const float* logits = (const float*)d_in[0];   // [1000,11]
    const float* reg    = (const float*)d_in[1];   // [1000,44]
    const float* props  = (const float*)d_in[2];   // [1000,4]
    float* out = (float*)d_out;                    // 600 floats

    // workspace layout: ms[10000] | nb[40000] | raw[40000]  (360 KB)
    float* ms  = (float*)d_ws;
    float* nb  = ms + M_CAND;
    float* raw = nb + M_CAND * 4;

    roih_decode_kernel<<<(M_CAND + 255) / 256, 256, 0, stream>>>(
        logits, reg, props, ms, nb, raw);
    roih_nms_kernel<<<1, NMS_THREADS, 0, stream>>>(ms, nb, raw, out);
}